// GeoDist_84619445666030
// MI455X (gfx1250) — compile-verified
//
#include <hip/hip_runtime.h>

// GCN forward: X@W0 -> propagate -> @W1 -> relu -> propagate -> @W2
// MI455X plan:
//  - GEMMs via V_WMMA_F32_16X16X4_F32 (BW-bound; fp32 kept, LDS-staged tiles).
//  - Propagation rebuilt as CSR gather (atomic-free): device-built CSR
//    (histogram + 2-level scan + cursor scatter), then one wave32 per node
//    accumulates dinv[i]*dinv[c]*H[c] in registers. Avoids 410M fp32 L2
//    atomics and two 102MB init passes of the naive scatter design.
//  - H (51MB) and P (51MB) are L2-resident (192MB L2), so gathers stay on-chip.

typedef __attribute__((ext_vector_type(2))) float v2f;
typedef __attribute__((ext_vector_type(8))) float v8f;

#define F_IN 128   // feature width of X / hidden

// ---------------- degree / normalization / CSR build ----------------

__global__ void k_zero_i(int* __restrict__ p, int n) {
  int i = blockIdx.x * blockDim.x + threadIdx.x;
  if (i < n) p[i] = 0;
}

__global__ void k_count(const int* __restrict__ row, int* __restrict__ cnt, int E) {
  int e = blockIdx.x * blockDim.x + threadIdx.x;
  if (e < E) atomicAdd(&cnt[row[e]], 1);
}

// dinv[i] = rsqrt(#row-edges + 1)   (+1 = self loop, matches reference segment_sum over r)
__global__ void k_dinv(const int* __restrict__ cnt, float* __restrict__ d, int n) {
  int i = blockIdx.x * blockDim.x + threadIdx.x;
  if (i < n) d[i] = __frsqrt_rn((float)cnt[i] + 1.0f);
}

// per-block inclusive scan (Hillis-Steele in LDS) + block totals
__global__ void k_scan_block(const int* __restrict__ cnt, int* __restrict__ incl,
                             int* __restrict__ bsum, int n) {
  __shared__ int s[256];
  int i = blockIdx.x * 256 + threadIdx.x;
  s[threadIdx.x] = (i < n) ? cnt[i] : 0;
  __syncthreads();
  for (int d = 1; d < 256; d <<= 1) {
    int t = (threadIdx.x >= d) ? s[threadIdx.x - d] : 0;
    __syncthreads();
    s[threadIdx.x] += t;
    __syncthreads();
  }
  if (i < n) incl[i] = s[threadIdx.x];
  if (threadIdx.x == 255) bsum[blockIdx.x] = s[255];
}

// serial exclusive scan of the ~391 block totals (trivial work)
__global__ void k_scan_sums(const int* __restrict__ bsum, int* __restrict__ bofs, int nb) {
  int acc = 0;
  for (int b = 0; b < nb; ++b) { bofs[b] = acc; acc += bsum[b]; }
}

// exclusive row start = inclusive - own + block offset
__global__ void k_start(const int* __restrict__ incl, const int* __restrict__ cnt,
                        const int* __restrict__ bofs, int* __restrict__ startA, int n) {
  int i = blockIdx.x * blockDim.x + threadIdx.x;
  if (i < n) startA[i] = incl[i] - cnt[i] + bofs[i >> 8];
}

// scatter column indices into the row's contiguous segment
__global__ void k_build(const int* __restrict__ row, const int* __restrict__ col,
                        const int* __restrict__ startA, int* __restrict__ cursor,
                        int* __restrict__ colsort, int E) {
  int e = blockIdx.x * blockDim.x + threadIdx.x;
  if (e < E) {
    int r = row[e];
    int pos = startA[r] + atomicAdd(&cursor[r], 1);
    colsort[pos] = col[e];
  }
}

// ---------------- GCN propagation (CSR gather, atomic-free) ----------------
// one wave32 per node; 4 features per lane; self-loop term = accumulator init.
__global__ void k_prop_csr(const float4* __restrict__ H, const float* __restrict__ dinv,
                           const int* __restrict__ startA, const int* __restrict__ cnt,
                           const int* __restrict__ colsort, float4* __restrict__ P, int n) {
  int gid = blockIdx.x * blockDim.x + threadIdx.x;
  int node = gid >> 5;
  int lane = gid & 31;
  if (node >= n) return;
  float di = dinv[node];
  float s2 = di * di;
  float4 h = H[(size_t)node * 32 + lane];
  float4 acc = make_float4(s2 * h.x, s2 * h.y, s2 * h.z, s2 * h.w);
  int j0 = startA[node];
  int j1 = j0 + cnt[node];
  for (int j = j0; j < j1; ++j) {       // uniform trip count per wave
    int c = colsort[j];                 // wave-uniform load (broadcast)
    float w = di * dinv[c];
    float4 v = H[(size_t)c * 32 + lane]; // coalesced 512B gather, L2-resident
    acc.x += w * v.x; acc.y += w * v.y; acc.z += w * v.z; acc.w += w * v.w;
  }
  P[(size_t)node * 32 + lane] = acc;
}

// ---------------- WMMA GEMM: Y[M,NCOLS] = X[M,128] @ W[128,NCOLS] ----------------
// 256 threads = 8 waves; block does 128 rows (16 rows/wave).
// W (128xNCOLS) staged once in LDS (stride NCOLS+8 -> half-wave K-phases hit
// disjoint bank groups); A tile 16x128 per wave in LDS (stride 132 ->
// conflict-free f32 frag reads, 16B-aligned float4 stores).

template <int NCOLS, bool RELU>
__global__ void k_gemm(const float* __restrict__ X, const float* __restrict__ W,
                       float* __restrict__ Y, int M) {
  constexpr int NP = NCOLS + 8;   // W LDS row stride (floats)
  constexpr int AP = 132;         // A LDS row stride (floats)
  constexpr int NT = NCOLS / 16;  // N tiles per wave

  extern __shared__ float smem[];
  float* Ws = smem;                                   // 128 * NP
  const int tid = threadIdx.x;
  const int wave = tid >> 5;
  const int lane = tid & 31;
  const int half = lane >> 4;
  const int lm = lane & 15;
  float* As = smem + 128 * NP + wave * 16 * AP;       // this wave's 16x128 tile

  // stage W cooperatively (float4, coalesced)
  for (int i = tid * 4; i < 128 * NCOLS; i += 256 * 4) {
    float4 w = *reinterpret_cast<const float4*>(W + i);
    int rr = i / NCOLS, cc = i % NCOLS;
    *reinterpret_cast<float4*>(Ws + rr * NP + cc) = w;
  }

  // stage this wave's A tile: one full row (32 lanes x float4) per iteration
  const int row0 = blockIdx.x * 128 + wave * 16;
  for (int r = 0; r < 16; ++r) {
    int gr = row0 + r;
    if (gr >= M) gr = M - 1;  // clamp tail; stores are guarded below
    *reinterpret_cast<float4*>(As + r * AP + lane * 4) =
        *reinterpret_cast<const float4*>(X + (size_t)gr * F_IN + lane * 4);
  }
  __syncthreads();

  v8f acc[NT];
#pragma unroll
  for (int n = 0; n < NT; ++n)
#pragma unroll
    for (int j = 0; j < 8; ++j) acc[n][j] = 0.0f;

  // K loop: 32 steps of V_WMMA_F32_16X16X4_F32 per N tile.
  // A frag 16x4: lanes 0-15 hold K={k0,k0+1}, lanes 16-31 hold K={k0+2,k0+3}.
  // B frag 4x16: mirror layout, N striped across lanes.
#pragma unroll 4
  for (int k0 = 0; k0 < 128; k0 += 4) {
    v2f a = *reinterpret_cast<const v2f*>(As + lm * AP + k0 + 2 * half);
    const float* wp = Ws + (k0 + 2 * half) * NP + lm;
#pragma unroll
    for (int n = 0; n < NT; ++n) {
      v2f b;
      b.x = wp[n * 16];
      b.y = wp[n * 16 + NP];
      acc[n] = __builtin_amdgcn_wmma_f32_16x16x4_f32(
          false, a, false, b, (short)0, acc[n], false, false);
    }
  }

  // C/D layout: VGPR j, lanes 0-15 -> row j, lanes 16-31 -> row j+8; col = lane%16
#pragma unroll
  for (int n = 0; n < NT; ++n) {
#pragma unroll
    for (int j = 0; j < 8; ++j) {
      int r = row0 + j + half * 8;
      if (r < M) {
        float v = acc[n][j];
        if (RELU) v = fmaxf(v, 0.0f);
        Y[(size_t)r * NCOLS + n * 16 + lm] = v;
      }
    }
  }
}

// ---------------- launch ----------------

extern "C" void kernel_launch(void* const* d_in, const int* in_sizes, int n_in,
                              void* d_out, int out_size, void* d_ws, size_t ws_size,
                              hipStream_t stream) {
  const float* X  = (const float*)d_in[0];   // [N,128]
  const int*   ei = (const int*)d_in[1];     // [2,E] int32
  const float* W0 = (const float*)d_in[2];   // [128,128]
  const float* W1 = (const float*)d_in[3];   // [128,128]
  const float* W2 = (const float*)d_in[4];   // [128,64]
  float* out = (float*)d_out;                // [N,64]

  const int N = in_sizes[0] / F_IN;          // 100000
  const int E = in_sizes[1] / 2;             // 1600000
  const int* rowp = ei;
  const int* colp = ei + E;

  // workspace layout
  char* ws = (char*)d_ws;
  float* dinv   = (float*)(ws);                    // N floats      @ 0
  int*   cnt    = (int*)(ws + (1u << 19));         // N ints        @ 0.5MB
  int*   startA = (int*)(ws + (1u << 20));         // N ints        @ 1.0MB
  int*   cursor = (int*)(ws + 3u * (1u << 19));    // N ints        @ 1.5MB
  int*   incl   = (int*)(ws + (1u << 21));         // N ints        @ 2.0MB
  int*   bsum   = (int*)(ws + 5u * (1u << 19));    // NB ints       @ 2.5MB
  int*   bofs   = (int*)(ws + 5u * (1u << 19) + 65536);
  int*   colsort= (int*)(ws + 3u * (1u << 20));    // E ints        @ 3.0MB (6.4MB)
  float* bufA   = (float*)(ws + (16u << 20));      // N*128 floats  @ 16MB (51.2MB)
  float* bufB   = bufA + (size_t)N * F_IN;         //               @ 67.2MB (51.2MB)

  const int T = 256;
  const int nb_N  = (N + T - 1) / T;        // also = #scan blocks
  const int nb_E  = (E + T - 1) / T;
  const int nb_NW = (N * 32 + T - 1) / T;   // wave per node
  const int nb_G  = (N + 127) / 128;        // gemm blocks

  const size_t sh128 = (size_t)(128 * (128 + 8) + 8 * 16 * 132) * sizeof(float); // ~137KB
  const size_t sh64  = (size_t)(128 * (64 + 8)  + 8 * 16 * 132) * sizeof(float); // ~105KB
  hipFuncSetAttribute((const void*)k_gemm<128, false>,
                      hipFuncAttributeMaxDynamicSharedMemorySize, (int)sh128);
  hipFuncSetAttribute((const void*)k_gemm<128, true>,
                      hipFuncAttributeMaxDynamicSharedMemorySize, (int)sh128);
  hipFuncSetAttribute((const void*)k_gemm<64, false>,
                      hipFuncAttributeMaxDynamicSharedMemorySize, (int)sh64);

  // ---- build normalization + CSR (cheap: ~3.2M int atomics, once per call) ----
  k_zero_i<<<nb_N, T, 0, stream>>>(cnt, N);
  k_zero_i<<<nb_N, T, 0, stream>>>(cursor, N);
  k_count<<<nb_E, T, 0, stream>>>(rowp, cnt, E);
  k_dinv<<<nb_N, T, 0, stream>>>(cnt, dinv, N);
  k_scan_block<<<nb_N, T, 0, stream>>>(cnt, incl, bsum, N);
  k_scan_sums<<<1, 1, 0, stream>>>(bsum, bofs, nb_N);
  k_start<<<nb_N, T, 0, stream>>>(incl, cnt, bofs, startA, N);
  k_build<<<nb_E, T, 0, stream>>>(rowp, colp, startA, cursor, colsort, E);

  // ---- layer 0: H0 = X @ W0 ----
  k_gemm<128, false><<<nb_G, T, sh128, stream>>>(X, W0, bufA, N);

  // ---- P1 = A_hat * H0 (atomic-free gather) ----
  k_prop_csr<<<nb_NW, T, 0, stream>>>((const float4*)bufA, dinv, startA, cnt,
                                      colsort, (float4*)bufB, N);

  // ---- H1 = relu(P1 @ W1) ----
  k_gemm<128, true><<<nb_G, T, sh128, stream>>>(bufB, W1, bufA, N);

  // ---- P2 = A_hat * H1 ----
  k_prop_csr<<<nb_NW, T, 0, stream>>>((const float4*)bufA, dinv, startA, cnt,
                                      colsort, (float4*)bufB, N);

  // ---- OUT = P2 @ W2 ----
  k_gemm<64, false><<<nb_G, T, sh64, stream>>>(bufB, W2, out, N);
}